// ProST_42769284333743
// MI455X (gfx1250) — compile-verified
//
#include <hip/hip_runtime.h>
#include <hip/hip_bf16.h>

// ProST forward projector for MI455X (gfx1250, wave32).
//
//  - Ray march is affine in t:  q(t) = p0' + t * dv'  (voxel scaling folded in).
//  - Per wave: 16 rays (adjacent detector pixels). Coordinates for each
//    16-sample x 16-ray tile come from three V_WMMA_F32_16X16X4_F32 using the
//    K=0 column for t*dv' and the K=1 column for 1*p0' (C = inline 0):
//        D[s][n] = t_s * dv'_i(n) + 1 * p0'_i
//    C/D layout: lane = ray (N), VGPR = sample (M); lanes 16-31 carry samples
//    m+8 -> at every gather step lanes 0-15 are 16 adjacent pixels (coalesced
//    x-voxel gathers). Volume (134 MB) fits in the 192 MB L2: L2-gather bound.
//  - Clamp-first trilinear via v_med3 (no canonicalize): when a coordinate is
//    clamped its fractional weight is 0, so results match the reference's
//    index-clamping exactly; upper-edge delta=0 prevents OOB reads.
//  - Corner addresses: base=(z<<16)+(y<<8)+x (2x v_lshl_add_u32) + 7 adds of
//    {dx,dy,dz} deltas.

typedef __attribute__((ext_vector_type(2))) float v2f;
typedef __attribute__((ext_vector_type(8))) float v8f;

#define DV 256   // volume depth
#define HV 256   // volume height
#define WV 256   // volume width
#define HD 256   // detector rows
#define WD 256   // detector cols
#define NB 2     // batch

template <bool GUARD>
__device__ __forceinline__ void prost_tile(
    const float* __restrict__ vol, float tstep, float dmin,
    int tt, int lane, int shalf, int K,
    v2f Bx, v2f By, v2f Bz, float& acc)
{
  // A: K=0 column = t for sample row m (lanes 0-15); K=1 column = 1.0 to pick
  // up p0' from B's K=1 row. Lanes 16-31 (K=2,3) multiply zero rows of B.
  const float tA = fmaf(tstep, (float)(tt * 16 + lane), dmin);
  const v2f A = { tA, 1.0f };
  const v8f Z8 = { 0.0f, 0.0f, 0.0f, 0.0f, 0.0f, 0.0f, 0.0f, 0.0f };

  v8f gX = __builtin_amdgcn_wmma_f32_16x16x4_f32(
      false, A, false, Bx, (short)0, Z8, false, false);
  v8f gY = __builtin_amdgcn_wmma_f32_16x16x4_f32(
      false, A, false, By, (short)0, Z8, false, false);
  v8f gZ = __builtin_amdgcn_wmma_f32_16x16x4_f32(
      false, A, false, Bz, (short)0, Z8, false, false);

  #pragma unroll
  for (int g = 0; g < 8; ++g) {
    if (!GUARD || (tt * 16 + g + shalf) < K) {
      // Clamp-first trilinear (equivalent to reference index clamping).
      const float gx = __builtin_amdgcn_fmed3f(gX[g], 0.0f, (float)(WV - 1));
      const float gy = __builtin_amdgcn_fmed3f(gY[g], 0.0f, (float)(HV - 1));
      const float gz = __builtin_amdgcn_fmed3f(gZ[g], 0.0f, (float)(DV - 1));
      const float fx = floorf(gx), fy = floorf(gy), fz = floorf(gz);
      const float xd = gx - fx, yd = gy - fy, zd = gz - fz;
      const int x0 = (int)fx, y0 = (int)fy, z0 = (int)fz;
      // Neighbor deltas: 0 at the upper edge (where the weight is 0 anyway).
      const unsigned dx = (x0 < WV - 1) ? 1u : 0u;
      const unsigned dy = (y0 < HV - 1) ? 256u : 0u;
      const unsigned dz = (z0 < DV - 1) ? 65536u : 0u;

      const unsigned i000 = ((unsigned)z0 << 16) + ((unsigned)y0 << 8) + (unsigned)x0;
      const unsigned i001 = i000 + dx;
      const unsigned i010 = i000 + dy;
      const unsigned i011 = i001 + dy;
      const unsigned i100 = i000 + dz;
      const unsigned i101 = i001 + dz;
      const unsigned i110 = i010 + dz;
      const unsigned i111 = i011 + dz;

      const float v000 = vol[i000], v001 = vol[i001];
      const float v010 = vol[i010], v011 = vol[i011];
      const float v100 = vol[i100], v101 = vol[i101];
      const float v110 = vol[i110], v111 = vol[i111];

      const float c00 = fmaf(xd, v001 - v000, v000);
      const float c01 = fmaf(xd, v011 - v010, v010);
      const float c10 = fmaf(xd, v101 - v100, v100);
      const float c11 = fmaf(xd, v111 - v110, v110);
      const float c0  = fmaf(yd, c01 - c00, c00);
      const float c1  = fmaf(yd, c11 - c10, c10);
      acc += fmaf(zd, c1 - c0, c0);
    }
  }
}

__global__ __launch_bounds__(256) void prost_fwd_kernel(
    const float* __restrict__ x,      // [B,1,D,H,W]
    const float* __restrict__ tmat,   // [B,3,4]
    const float* __restrict__ corner, // [B,8,4]
    const float* __restrict__ param,  // [src, det, pix_spacing, step]
    const int*   __restrict__ Kp,     // scalar K
    float* __restrict__ out)          // [B,1,HD,WD]
{
  const int lane  = threadIdx.x & 31;
  const int wave  = threadIdx.x >> 5;
  const int b     = blockIdx.z;
  const int h     = blockIdx.y;
  const int wbase = blockIdx.x * 128 + wave * 16;
  const int n     = lane & 15;          // ray slot within wave
  const int w     = wbase + n;          // detector column
  const bool lo   = (lane < 16);

  const float src = param[0];
  const float det = param[1];
  const float pix = param[2];
  const int   K   = *Kp;

  // Pose rows (uniform per block -> scalarized).
  const float* T = tmat + b * 12;
  const float T00=T[0], T01=T[1], T02=T[2],  T03=T[3];
  const float T10=T[4], T11=T[5], T12=T[6],  T13=T[7];
  const float T20=T[8], T21=T[9], T22=T[10], T23=T[11];

  // dist_min / dist_max from the 8 transformed corners (cheap, uniform).
  float dmin = 1e30f, dmax = -1e30f;
  const float* cp = corner + b * 32;
  #pragma unroll
  for (int c = 0; c < 8; ++c) {
    const float c0 = cp[c*4+0], c1 = cp[c*4+1], c2 = cp[c*4+2], c3 = cp[c*4+3];
    const float ex = T00*c0 + T01*c1 + T02*c2 + T03*c3;
    const float ey = T10*c0 + T11*c1 + T12*c2 + T13*c3;
    const float ez = T20*c0 + T21*c1 + T22*c2 + T23*c3 - src;
    const float d  = sqrtf(ex*ex + ey*ey + ez*ez);
    dmin = fminf(dmin, d);
    dmax = fmaxf(dmax, d);
  }

  // Per-ray direction dir = (px, py, -det)/|.|;
  // dv' = (T*(dir,0)) * 127.5 ; p0' = ((T*(0,0,src,1)) + 1) * 127.5
  const float px  = ((float)w - (float)(WD - 1) * 0.5f) * pix;
  const float py  = ((float)h - (float)(HD - 1) * 0.5f) * pix;
  const float inv = 1.0f / sqrtf(px*px + py*py + det*det);
  const float SC  = 127.5f;  // (dim-1)/2 for all three 256 dims
  const float dvx = (T00*px + T01*py - T02*det) * inv * SC;
  const float dvy = (T10*px + T11*py - T12*det) * inv * SC;
  const float dvz = (T20*px + T21*py - T22*det) * inv * SC;
  const float p0x = (T02*src + T03 + 1.0f) * SC;
  const float p0y = (T12*src + T13 + 1.0f) * SC;
  const float p0z = (T22*src + T23 + 1.0f) * SC;

  // B: K=0 row = dv' (per-ray), K=1 row = p0' (broadcast); rows K=2,3 zero.
  const v2f Bx = { lo ? dvx : 0.0f, lo ? p0x : 0.0f };
  const v2f By = { lo ? dvy : 0.0f, lo ? p0y : 0.0f };
  const v2f Bz = { lo ? dvz : 0.0f, lo ? p0z : 0.0f };

  const float* vol   = x + (size_t)b * (DV * HV * WV);
  const float  tstep = (dmax - dmin) / (float)(K - 1);
  const int    shalf = (lane >> 4) << 3;  // 0 (lanes 0-15) or 8 (lanes 16-31)

  float acc = 0.0f;

  // Guard-free main tiles (K % 16 == 0 covers the reference config K=128).
  const int fullTiles = K >> 4;
  for (int tt = 0; tt < fullTiles; ++tt) {
    prost_tile<false>(vol, tstep, dmin, tt, lane, shalf, K, Bx, By, Bz, acc);
  }
  // Guarded tail tile (uniform branch; EXEC all-ones around the WMMAs).
  if (K & 15) {
    prost_tile<true>(vol, tstep, dmin, fullTiles, lane, shalf, K,
                     Bx, By, Bz, acc);
  }

  // Combine the two half-wave partial sums for each ray; lanes 0-15 store.
  const float tot = acc + __shfl_xor(acc, 16, 32);
  if (lo) {
    out[(size_t)(b * HD + h) * WD + w] = tot;
  }
}

extern "C" void kernel_launch(void* const* d_in, const int* in_sizes, int n_in,
                              void* d_out, int out_size, void* d_ws, size_t ws_size,
                              hipStream_t stream) {
  (void)in_sizes; (void)n_in; (void)out_size; (void)d_ws; (void)ws_size;
  const float* x      = (const float*)d_in[0];
  // d_in[1] = y (detector image): only its shape matters, unused on device.
  const float* tmat   = (const float*)d_in[2];
  const float* corner = (const float*)d_in[3];
  const float* param  = (const float*)d_in[4];
  const int*   Kp     = (const int*)  d_in[5];
  float*       out    = (float*)d_out;

  dim3 grid(WD / 128, HD, NB);  // 2 x 256 x 2 blocks, 128 pixels per block
  prost_fwd_kernel<<<grid, 256, 0, stream>>>(x, tmat, corner, param, Kp, out);
}